// GConvNetwork_85727547228592
// MI455X (gfx1250) — compile-verified
//
#include <hip/hip_runtime.h>
#include <hip/hip_bf16.h>

typedef __attribute__((ext_vector_type(2))) float v2f;
typedef __attribute__((ext_vector_type(8))) float v8f;

#define EMB 128

// ---------------------------------------------------------------------------
// Degree / normalization
// ---------------------------------------------------------------------------
__global__ void deg_init_kernel(float* __restrict__ deg, int n) {
    int i = blockIdx.x * blockDim.x + threadIdx.x;
    if (i < n) deg[i] = 1.0f;  // self-loop contributes 1 to every node's degree
}

__global__ void deg_count_kernel(const int* __restrict__ dst, float* __restrict__ deg, int e) {
    int i = blockIdx.x * blockDim.x + threadIdx.x;
    if (i < e) atomicAdd(&deg[dst[i]], 1.0f);
}

__global__ void deg_to_dinv_kernel(float* __restrict__ deg, int n) {
    int i = blockIdx.x * blockDim.x + threadIdx.x;
    if (i < n) deg[i] = __frsqrt_rn(deg[i]);  // deg >= 1 always (self-loops)
}

// ---------------------------------------------------------------------------
// H = X @ W via V_WMMA_F32_16X16X4_F32.
// One wave per 16x16 output tile. Block = 256 threads = 8 waves = the 8
// column tiles of one 16-row stripe. K (=128) is consumed 4 at a time.
//
// fp32 A (16x4) layout:  lanes 0-15 hold {K=0,K=1}, lanes 16-31 hold {K=2,K=3}
// fp32 B (4x16) layout:  mirrored (lane = column, half selects K pair)
// fp32 C/D (16x16):      VGPR r -> row r (lanes 0-15), row r+8 (lanes 16-31)
// ---------------------------------------------------------------------------
__global__ __launch_bounds__(256) void gemm_wmma_kernel(
    const float* __restrict__ X, const float* __restrict__ W,
    float* __restrict__ H, int nrows) {
    const int wave = threadIdx.x >> 5;   // 0..7 : column tile
    const int lane = threadIdx.x & 31;
    const int half = lane >> 4;          // 0 or 1
    const int l    = lane & 15;
    const int row0 = blockIdx.x * 16;
    if (row0 >= nrows) return;           // uniform per wave -> EXEC stays all-1s
    const int col0 = wave * 16;

    int arow = row0 + l;
    if (arow >= nrows) arow = nrows - 1;           // clamp (only ragged last tile)
    const float* __restrict__ xrow = X + (size_t)arow * EMB;

    v8f c = {};
    #pragma unroll 4
    for (int k = 0; k < EMB; k += 4) {
        const int kk = k + half * 2;
        v2f a, b;
        a.x = xrow[kk];
        a.y = xrow[kk + 1];
        b.x = W[(size_t)kk * EMB + col0 + l];
        b.y = W[(size_t)(kk + 1) * EMB + col0 + l];
        c = __builtin_amdgcn_wmma_f32_16x16x4_f32(
            false, a, false, b, (short)0, c, false, false);
    }

    const int col = col0 + l;
    float* __restrict__ base = H + (size_t)(row0 + half * 8) * EMB + col;
    if (row0 + 16 <= nrows) {
        // full tile: branch-free store epilogue (uniform predicate)
        #pragma unroll
        for (int r = 0; r < 8; ++r)
            base[(size_t)r * EMB] = c[r];
    } else {
        #pragma unroll
        for (int r = 0; r < 8; ++r) {
            const int row = row0 + r + half * 8;
            if (row < nrows) H[(size_t)row * EMB + col] = c[r];
        }
    }
}

// ---------------------------------------------------------------------------
// Edge scatter: one wave per edge, each lane moves a float4 (32*4 = 128).
// acc[dst] += h[src] * dinv[src]*dinv[dst]  via global_atomic_add_f32.
// acc and h both fit in MI455X's 192MB L2, so atomics resolve on-chip.
// ---------------------------------------------------------------------------
__global__ __launch_bounds__(256) void scatter_kernel(
    const float* __restrict__ H, const int* __restrict__ src,
    const int* __restrict__ dst, const float* __restrict__ dinv,
    float* __restrict__ acc, int nedges) {
    const int wavesPerBlock = blockDim.x >> 5;
    const int lane = threadIdx.x & 31;
    int e = blockIdx.x * wavesPerBlock + (threadIdx.x >> 5);
    const int stride = gridDim.x * wavesPerBlock;
    for (; e < nedges; e += stride) {
        const int s = src[e];
        const int d = dst[e];
        const float norm = dinv[s] * dinv[d];
        const float4 v = ((const float4*)(H + (size_t)s * EMB))[lane];
        float* ap = acc + (size_t)d * EMB + lane * 4;
        atomicAdd(ap + 0, v.x * norm);
        atomicAdd(ap + 1, v.y * norm);
        atomicAdd(ap + 2, v.z * norm);
        atomicAdd(ap + 3, v.w * norm);
    }
}

// ---------------------------------------------------------------------------
// out = relu(acc + h * dinv^2 (self-loop) + bias).  out may alias H.
// ---------------------------------------------------------------------------
__global__ __launch_bounds__(256) void finalize_kernel(
    const float* __restrict__ acc, const float* __restrict__ H,
    const float* __restrict__ dinv, const float* __restrict__ bias,
    float* __restrict__ out, int n) {
    const int idx = blockIdx.x * blockDim.x + threadIdx.x;
    if (idx < n * EMB) {
        const int node = idx >> 7;
        const int d    = idx & (EMB - 1);
        const float di = dinv[node];
        float v = acc[idx] + H[idx] * di * di + bias[d];
        out[idx] = v > 0.0f ? v : 0.0f;
    }
}

// ---------------------------------------------------------------------------
// Launch
// ---------------------------------------------------------------------------
extern "C" void kernel_launch(void* const* d_in, const int* in_sizes, int n_in,
                              void* d_out, int out_size, void* d_ws, size_t ws_size,
                              hipStream_t stream) {
    const float* x0 = (const float*)d_in[0];
    const float* W1 = (const float*)d_in[1];
    const float* b1 = (const float*)d_in[2];
    const float* W2 = (const float*)d_in[3];
    const float* b2 = (const float*)d_in[4];
    const float* W3 = (const float*)d_in[5];
    const float* b3 = (const float*)d_in[6];
    const int*   ei = (const int*)d_in[7];

    const int N = in_sizes[0] / EMB;      // 50000
    const int E = in_sizes[7] / 2;        // 600000
    const int* src = ei;
    const int* dst = ei + E;

    // workspace layout: dinv | bufA | bufB | acc
    float* dinv = (float*)d_ws;
    size_t off = ((size_t)N + 63) & ~(size_t)63;
    float* bufA = dinv + off;
    float* bufB = bufA + (size_t)N * EMB;
    float* acc  = bufB + (size_t)N * EMB;
    float* out  = (float*)d_out;

    const size_t ndBytes = (size_t)N * EMB * sizeof(float);
    const int tilesN = (N + 15) / 16;
    const int elemBlocks = (N * EMB + 255) / 256;
    const int edgeWaveBlocks = (E + 7) / 8;     // 8 waves/block, 1 wave/edge

    // normalization
    deg_init_kernel<<<(N + 255) / 256, 256, 0, stream>>>(dinv, N);
    deg_count_kernel<<<(E + 255) / 256, 256, 0, stream>>>(dst, dinv, E);
    deg_to_dinv_kernel<<<(N + 255) / 256, 256, 0, stream>>>(dinv, N);

    // ---- layer 1: x0 -> bufA
    gemm_wmma_kernel<<<tilesN, 256, 0, stream>>>(x0, W1, bufA, N);
    hipMemsetAsync(acc, 0, ndBytes, stream);
    scatter_kernel<<<edgeWaveBlocks, 256, 0, stream>>>(bufA, src, dst, dinv, acc, E);
    finalize_kernel<<<elemBlocks, 256, 0, stream>>>(acc, bufA, dinv, b1, bufA, N);

    // ---- layer 2: bufA -> bufB
    gemm_wmma_kernel<<<tilesN, 256, 0, stream>>>(bufA, W2, bufB, N);
    hipMemsetAsync(acc, 0, ndBytes, stream);
    scatter_kernel<<<edgeWaveBlocks, 256, 0, stream>>>(bufB, src, dst, dinv, acc, E);
    finalize_kernel<<<elemBlocks, 256, 0, stream>>>(acc, bufB, dinv, b2, bufB, N);

    // ---- layer 3: bufB -> d_out
    gemm_wmma_kernel<<<tilesN, 256, 0, stream>>>(bufB, W3, bufA, N);
    hipMemsetAsync(acc, 0, ndBytes, stream);
    scatter_kernel<<<edgeWaveBlocks, 256, 0, stream>>>(bufA, src, dst, dinv, acc, E);
    finalize_kernel<<<elemBlocks, 256, 0, stream>>>(acc, bufA, dinv, b3, out, N);
}